// GIN_57071525429595
// MI455X (gfx1250) — compile-verified
//
#include <hip/hip_runtime.h>

#define NN   100000      // nodes
#define NE   1600000     // edges
#define NG   512         // graphs
#define HDIM 128
#define POOLW 384

typedef __attribute__((ext_vector_type(2))) float v2f;
typedef __attribute__((ext_vector_type(8))) float v8f;

// ---------------- utility kernels ----------------

__global__ __launch_bounds__(256) void zero_k(float* __restrict__ p, long n) {
    long i = (long)blockIdx.x * blockDim.x + threadIdx.x;
    if (i < n) p[i] = 0.0f;
}

__global__ __launch_bounds__(256) void copy4_k(const float4* __restrict__ s, float4* __restrict__ d, long n4) {
    long i = (long)blockIdx.x * blockDim.x + threadIdx.x;
    if (i < n4) d[i] = s[i];
}

// y[dst[e]] += x[src[e]] ; one wave per edge, each lane owns 4 of 128 feats
__global__ __launch_bounds__(256) void scatter_add_k(const float* __restrict__ x,
                                                     const int* __restrict__ src,
                                                     const int* __restrict__ dst,
                                                     float* __restrict__ y) {
    long tid = (long)blockIdx.x * blockDim.x + threadIdx.x;
    long e = tid >> 5;
    if (e >= NE) return;
    int lane = (int)(tid & 31);
    int s = src[e], d = dst[e];
    float4 v = *(const float4*)(x + (long)s * HDIM + lane * 4);
    float* yp = y + (long)d * HDIM + lane * 4;
    atomicAdd(yp + 0, v.x);
    atomicAdd(yp + 1, v.y);
    atomicAdd(yp + 2, v.z);
    atomicAdd(yp + 3, v.w);
}

// pack W[K,N] row-major into k-pair-interleaved v2f: Wp[kp*N+n] = {W[2kp][n], W[2kp+1][n]}
__global__ __launch_bounds__(256) void repack_k(const float* __restrict__ W, v2f* __restrict__ Wp,
                                                int K, int N) {
    int i = blockIdx.x * blockDim.x + threadIdx.x;
    if (i >= (K / 2) * N) return;
    int kp = i / N, n = i - kp * N;
    v2f t;
    t.x = W[(2 * kp) * N + n];
    t.y = W[(2 * kp + 1) * N + n];
    Wp[i] = t;
}

// ---------------- WMMA GEMM, K=N=128, weights in LDS as v2f k-pairs ----------------
// C[M,128] = act(A)[M,128] @ W[128,128] + b ; act = identity or fused BN+ReLU on A.
// 4 waves/block; each wave computes a 16x128 strip with 8 f32 16x16x4 accumulators.
// B fragment {W[k][n], W[k+1][n]} is one aligned ds_load_b64 (kills mov traffic).
template <int BNFUSE>
__global__ __launch_bounds__(128) void gemm128_wmma_k(const float* __restrict__ A,
                                                      const float* __restrict__ W,
                                                      const float* __restrict__ bias,
                                                      const float* __restrict__ bnss,
                                                      float* __restrict__ C,
                                                      int M, int do_relu) {
    __shared__ v2f sW2[64 * 144];   // 64 k-pairs x 128 cols, stride 144 -> halves hit disjoint banks
    for (int i = threadIdx.x; i < 64 * 128; i += 128) {
        int kp = i >> 7, n = i & 127;
        v2f t;
        t.x = W[(2 * kp) * HDIM + n];
        t.y = W[(2 * kp + 1) * HDIM + n];
        sW2[kp * 144 + n] = t;
    }
    __syncthreads();

    const int wave = threadIdx.x >> 5;
    const int lane = threadIdx.x & 31;
    const int ln = lane & 15;          // col-in-tile / A-frag row
    const int hi = lane >> 4;          // K-pair select (A/B), M+8 half (C)
    const int row_base = (blockIdx.x * 4 + wave) * 16;
    if (row_base >= M) return;         // wave-uniform; after the only barrier

    const float* arow = A + (long)(row_base + ln) * HDIM;
    v8f acc[8] = {};

    for (int k0 = 0; k0 < HDIM; k0 += 4) {
        const int c = k0 + 2 * hi;                     // even
        v2f av = *(const v2f*)(arow + c);              // A 16x4 frag: one b64 load
        if constexpr (BNFUSE) {
            v2f sc = *(const v2f*)(bnss + c);
            v2f sh = *(const v2f*)(bnss + 128 + c);
            av.x = fmaxf(av.x * sc.x + sh.x, 0.0f);
            av.y = fmaxf(av.y * sc.y + sh.y, 0.0f);
        }
        const v2f* wrow = sW2 + (c >> 1) * 144;
#pragma unroll
        for (int t = 0; t < 8; t++) {
            v2f bv = wrow[t * 16 + ln];                // one ds_load_b64
            acc[t] = __builtin_amdgcn_wmma_f32_16x16x4_f32(
                false, av, false, bv, (short)0, acc[t], false, false);
        }
    }

#pragma unroll
    for (int t = 0; t < 8; t++) {
        const int col = t * 16 + ln;
        const float bb = bias[col];
#pragma unroll
        for (int r = 0; r < 8; r++) {
            float v = acc[t][r] + bb;
            if (do_relu) v = fmaxf(v, 0.0f);
            C[(long)(row_base + r + 8 * hi) * HDIM + col] = v;
        }
    }
}

// ---------------- WMMA GEMM, generic K, N multiple of 128, packed W from global ----------------
__global__ __launch_bounds__(128) void gemm_glob_wmma_k(const float* __restrict__ A,
                                                        const v2f* __restrict__ Wp,
                                                        const float* __restrict__ bias,
                                                        float* __restrict__ C,
                                                        int M, int N, int K, int do_relu) {
    const int wave = threadIdx.x >> 5;
    const int lane = threadIdx.x & 31;
    const int ln = lane & 15;
    const int hi = lane >> 4;
    const int row_base = (blockIdx.x * 4 + wave) * 16;
    if (row_base >= M) return;

    const float* arow = A + (long)(row_base + ln) * K;

    for (int nc = 0; nc < N; nc += 128) {
        v8f acc[8] = {};
        for (int k0 = 0; k0 < K; k0 += 4) {
            const int c = k0 + 2 * hi;
            v2f av = *(const v2f*)(arow + c);
            const v2f* wrow = Wp + (long)(c >> 1) * N + nc;
#pragma unroll
            for (int t = 0; t < 8; t++) {
                v2f bv = wrow[t * 16 + ln];            // one global b64 load
                acc[t] = __builtin_amdgcn_wmma_f32_16x16x4_f32(
                    false, av, false, bv, (short)0, acc[t], false, false);
            }
        }
#pragma unroll
        for (int t = 0; t < 8; t++) {
            const int col = nc + t * 16 + ln;
            const float bb = bias[col];
#pragma unroll
            for (int r = 0; r < 8; r++) {
                float v = acc[t][r] + bb;
                if (do_relu) v = fmaxf(v, 0.0f);
                C[(long)(row_base + r + 8 * hi) * N + col] = v;
            }
        }
    }
}

// ---------------- BN helpers ----------------

// per-column sum / sumsq of h[M,128] into stats[0..127]=sum, stats[128..255]=sumsq
__global__ __launch_bounds__(128) void colstats_k(const float* __restrict__ h,
                                                  float* __restrict__ stats, int M) {
    const int c = threadIdx.x;
    const int chunk = (M + gridDim.x - 1) / gridDim.x;
    long r0 = (long)blockIdx.x * chunk;
    long r1 = r0 + chunk; if (r1 > M) r1 = M;
    float s = 0.0f, q = 0.0f;
    for (long r = r0; r < r1; r++) {
        float v = h[r * HDIM + c];
        s += v; q += v * v;
    }
    atomicAdd(&stats[c], s);
    atomicAdd(&stats[HDIM + c], q);
}

// ss[0..127] = g*rsqrt(var+eps), ss[128..255] = beta - mu*scale
__global__ __launch_bounds__(128) void scaleshift_k(const float* __restrict__ stats,
                                                    const float* __restrict__ g,
                                                    const float* __restrict__ be,
                                                    float* __restrict__ ss) {
    int c = threadIdx.x;
    const float invM = 1.0f / (float)NN;
    float mu = stats[c] * invM;
    float var = stats[HDIM + c] * invM - mu * mu;
    float sc = g[c] * rsqrtf(var + 1e-5f);
    ss[c] = sc;
    ss[HDIM + c] = be[c] - mu * sc;
}

// ---------------- pooling & head ----------------

__global__ __launch_bounds__(256) void pool_k(const float* __restrict__ h,
                                              const int* __restrict__ batch,
                                              float* __restrict__ pool, int off) {
    long i = (long)blockIdx.x * blockDim.x + threadIdx.x;
    if (i >= (long)NN * HDIM) return;
    int n = (int)(i >> 7), c = (int)(i & 127);
    atomicAdd(&pool[(long)batch[n] * POOLW + off + c], h[i]);
}

// out[512,10] = A[512,384] @ W[384,10] + b   (tiny; VALU)
__global__ __launch_bounds__(256) void lin2_k(const float* __restrict__ A,
                                              const float* __restrict__ W,
                                              const float* __restrict__ b,
                                              float* __restrict__ out) {
    int i = blockIdx.x * blockDim.x + threadIdx.x;
    if (i >= NG * 10) return;
    int g = i / 10, c = i % 10;
    float s = b[c];
    for (int k = 0; k < POOLW; k++) s += A[(long)g * POOLW + k] * W[k * 10 + c];
    out[i] = s;
}

// ---------------- launch ----------------

extern "C" void kernel_launch(void* const* d_in, const int* in_sizes, int n_in,
                              void* d_out, int out_size, void* d_ws, size_t ws_size,
                              hipStream_t stream) {
    const float* x   = (const float*)d_in[0];
    const int*   src = (const int*)d_in[1];
    const int*   dst = src + NE;
    const int*   bat = (const int*)d_in[2];

    // conv params start at index 3, 6 tensors per conv.
    // sorted-key flatten -> (W1,W2,b1,b2,beta,g); insertion order -> (W1,b1,g,beta,W2,b2)
    const bool sorted_order = (in_sizes[4] == HDIM * HDIM);
    const float *W1[3], *B1[3], *G[3], *BE[3], *W2[3], *B2[3];
    for (int l = 0; l < 3; l++) {
        const int base = 3 + 6 * l;
        if (sorted_order) {
            W1[l] = (const float*)d_in[base + 0];
            W2[l] = (const float*)d_in[base + 1];
            B1[l] = (const float*)d_in[base + 2];
            B2[l] = (const float*)d_in[base + 3];
            BE[l] = (const float*)d_in[base + 4];
            G[l]  = (const float*)d_in[base + 5];
        } else {
            W1[l] = (const float*)d_in[base + 0];
            B1[l] = (const float*)d_in[base + 1];
            G[l]  = (const float*)d_in[base + 2];
            BE[l] = (const float*)d_in[base + 3];
            W2[l] = (const float*)d_in[base + 4];
            B2[l] = (const float*)d_in[base + 5];
        }
    }
    const float* lin1W = (const float*)d_in[21];
    const float* lin1b = (const float*)d_in[22];
    const float* lin2W = (const float*)d_in[23];
    const float* lin2b = (const float*)d_in[24];

    // workspace layout (floats)
    float* bufA  = (float*)d_ws;                        // [NN,128]
    float* bufB  = bufA + (long)NN * HDIM;              // [NN,128]
    float* pool  = bufB + (long)NN * HDIM;              // [512,384]
    float* l1o   = pool + (long)NG * POOLW;             // [512,384]
    float* stats = l1o + (long)NG * POOLW;              // [256]
    float* ss    = stats + 256;                         // [256]
    v2f*   wp    = (v2f*)(ss + 256);                    // lin1W packed: [192*384] v2f

    const long nodeElems = (long)NN * HDIM;             // 12.8M
    const int gemmGrid = (NN + 63) / 64;                // 1563 blocks (4 waves x 16 rows)

    zero_k<<<(NG * POOLW + 255) / 256, 256, 0, stream>>>(pool, (long)NG * POOLW);

    const float* xin = x;
    float* P = bufA;
    float* Q = bufB;
    for (int l = 0; l < 3; l++) {
        // y = x + segment_sum(x[src], dst)  -> P
        copy4_k<<<(int)((nodeElems / 4 + 255) / 256), 256, 0, stream>>>(
            (const float4*)xin, (float4*)P, nodeElems / 4);
        scatter_add_k<<<(int)(((long)NE * 32 + 255) / 256), 256, 0, stream>>>(xin, src, dst, P);
        // h = y @ W1 + b1  -> Q
        gemm128_wmma_k<0><<<gemmGrid, 128, 0, stream>>>(P, W1[l], B1[l], nullptr, Q, NN, 0);
        // training-mode batch stats + scale/shift
        zero_k<<<1, 256, 0, stream>>>(stats, 256);
        colstats_k<<<512, 128, 0, stream>>>(Q, stats, NN);
        scaleshift_k<<<1, 128, 0, stream>>>(stats, G[l], BE[l], ss);
        // h_out = relu( relu(BN(h)) @ W2 + b2 )  -> P   (BN+ReLU fused into A-fragment load)
        gemm128_wmma_k<1><<<gemmGrid, 128, 0, stream>>>(Q, W2[l], B2[l], ss, P, NN, 1);
        // global_add_pool into concat slot
        pool_k<<<(int)((nodeElems + 255) / 256), 256, 0, stream>>>(P, bat, pool, l * HDIM);
        xin = P;
        float* tmp = P; P = Q; Q = tmp;
    }

    // head: relu(pool @ lin1W + lin1b) @ lin2W + lin2b
    repack_k<<<((POOLW / 2) * POOLW + 255) / 256, 256, 0, stream>>>(lin1W, wp, POOLW, POOLW);
    gemm_glob_wmma_k<<<NG / 64, 128, 0, stream>>>(pool, wp, lin1b, l1o, NG, POOLW, POOLW, 1);
    lin2_k<<<(NG * 10 + 255) / 256, 256, 0, stream>>>(l1o, lin2W, lin2b, (float*)d_out);
}